// got_total_wing_loss_26388279067065
// MI455X (gfx1250) — compile-verified
//
#include <hip/hip_runtime.h>

// Wing-loss + covariance wing-loss for [B,21,3] joints on gfx1250.
// One batch per wave32. The 21x21 Gram matrices M*M^T (K=3) run on
// V_WMMA_F32_16X16X4_F32 (K=4, zero-padded). For a symmetric product the f32
// A(16x4) and B(4x16) per-lane layouts coincide, so one 2-VGPR operand serves
// both sides. D = Mp*Mp^T - Mg*Mg^T is built by chaining two WMMAs with a
// VALU-negated gt B-operand. Wing's log is raw v_log_f32 (arg >= 1.0, so no
// denorm fixup needed); TRANS co-executes with WMMA. Deterministic 2-stage
// reduction, no atomics.

typedef __attribute__((ext_vector_type(2))) float v2f;
typedef __attribute__((ext_vector_type(8))) float v8f;

#define WING_W       10.0f
#define WING_NEG_C   7.91759469228055f    // -C = 10*(ln6 - 1)
#define WING_W_LN2   6.93147180559945f    // W * ln(2): wing = W*ln(x) = (W*ln2)*log2(x)

__device__ __forceinline__ float fast_log2_ge1(float x) {
  // x >= 1.0 guaranteed: raw v_log_f32, skip libm denormal-range fixup.
#if __has_builtin(__builtin_amdgcn_logf)
  return __builtin_amdgcn_logf(x);
#else
  return __log2f(x);
#endif
}

__device__ __forceinline__ float wing_abs(float x) {
  float a  = __builtin_fabsf(x);
  float lg = WING_W_LN2 * fast_log2_ge1(__builtin_fmaf(a, 0.5f, 1.0f)); // W*log(1+a/EPS), EPS=2
  float ln = a + WING_NEG_C;
  return a < WING_W ? lg : ln;
}

__device__ __forceinline__ v8f wmma_f32(v2f a, v2f b, v8f c) {
  // D = A(16x4) * B(4x16) + C
  return __builtin_amdgcn_wmma_f32_16x16x4_f32(false, a, false, b, (short)0, c,
                                               false, false);
}

__device__ __forceinline__ float tile_wing_sum(v8f s) {
  float t = 0.0f;
#pragma unroll
  for (int i = 0; i < 8; ++i) t += wing_abs(0.5f * s[i]);  // cov = Gram/2
  return t;
}

__global__ __launch_bounds__(256) void wing_total_kernel(
    const float* __restrict__ jo, const float* __restrict__ jg,
    float* __restrict__ partial, int nbatch) {
  const int lane     = threadIdx.x & 31;
  const int widx     = threadIdx.x >> 5;
  const int nwaveblk = blockDim.x >> 5;
  const int gwave    = blockIdx.x * nwaveblk + widx;
  const int nwaves   = gridDim.x * nwaveblk;
  const int lo       = lane & 15;
  const bool hi      = lane >= 16;

  float acc = 0.0f;

  for (int b = gwave; b < nbatch; b += nwaves) {
    const float* pb = jo + (size_t)b * 63u;
    const float* gb = jg + (size_t)b * 63u;
    if (b + nwaves < nbatch) {  // prefetch next grid-stride batch (global_prefetch_b8)
      __builtin_prefetch(pb + (size_t)nwaves * 63u, 0, 3);
      __builtin_prefetch(gb + (size_t)nwaves * 63u, 0, 3);
    }

    // Lane L (<21) owns joint L: 3 consecutive floats per tensor, coalesced.
    float px = 0.f, py = 0.f, pz = 0.f, gx = 0.f, gy = 0.f, gz = 0.f;
    if (lane < 21) {
      px = pb[3 * lane + 0]; py = pb[3 * lane + 1]; pz = pb[3 * lane + 2];
      gx = gb[3 * lane + 0]; gy = gb[3 * lane + 1]; gz = gb[3 * lane + 2];
    }

    // Row-mean removal (mean over the 3 observations). Lanes >=21 stay zero.
    const float third = 1.0f / 3.0f;
    float pm = (px + py + pz) * third;
    float gm = (gx + gy + gz) * third;
    float mpx = px - pm, mpy = py - pm, mpz = pz - pm;
    float mgx = gx - gm, mgy = gy - gm, mgz = gz - gm;

    // Build WMMA operands.
    // f32 A 16x4 layout: lanes 0-15 {V0=K0,V1=K1}, lanes 16-31 {V0=K2,V1=K3(=0 pad)}.
    // B 4x16 mirrors it, so one register pair = both A(row-tile) and B(col-tile).
    float pz_lo = __shfl(mpz, lo, 32);        // z of joint (lane%16)      -> tile0 hi half
    float gz_lo = __shfl(mgz, lo, 32);
    float px_hi = __shfl(mpx, 16 + lo, 32);   // x/y of joint 16+(lane%16) -> tile1 lo half
    float py_hi = __shfl(mpy, 16 + lo, 32);
    float gx_hi = __shfl(mgx, 16 + lo, 32);
    float gy_hi = __shfl(mgy, 16 + lo, 32);

    v2f p0, p1, g0, g1, n0, n1;
    p0.x = hi ? pz_lo : mpx;  p0.y = hi ? 0.0f : mpy;   // rows/cols 0-15
    g0.x = hi ? gz_lo : mgx;  g0.y = hi ? 0.0f : mgy;
    p1.x = hi ? mpz  : px_hi; p1.y = hi ? 0.0f : py_hi; // rows/cols 16-31 (>=21 zero)
    g1.x = hi ? mgz  : gx_hi; g1.y = hi ? 0.0f : gy_hi;
    n0.x = -g0.x; n0.y = -g0.y;                         // f32 WMMA lacks B-neg: negate in VALU
    n1.x = -g1.x; n1.y = -g1.y;

    const v8f zc = {};
    float covsum;
    {   // tile (0:15, 0:15) — all 256 entries valid
      v8f s = wmma_f32(g0, n0, zc);   // -Mg Mg^T
      s     = wmma_f32(p0, p0, s);    // +Mp Mp^T
      covsum = tile_wing_sum(s);
    }
    {   // tile (0:15, 16:31) — symmetric partner counted x2; padded cols give wing(0)=0
      v8f s = wmma_f32(g0, n1, zc);
      s     = wmma_f32(p0, p1, s);
      covsum += 2.0f * tile_wing_sum(s);
    }
    {   // tile (16:31, 16:31)
      v8f s = wmma_f32(g1, n1, zc);
      s     = wmma_f32(p1, p1, s);
      covsum += tile_wing_sum(s);
    }

    // Point wing loss: 21 joint rows + centroid row, 3 coords each (66 terms).
    float dx = px - gx, dy = py - gy, dz = pz - gz;     // zero for lanes >= 21
    float psum = wing_abs(dx) + wing_abs(dy) + wing_abs(dz);

    float sx = dx, sy = dy, sz = dz;                    // centroid needs wave sum
#pragma unroll
    for (int m = 16; m > 0; m >>= 1) {
      sx += __shfl_xor(sx, m, 32);
      sy += __shfl_xor(sy, m, 32);
      sz += __shfl_xor(sz, m, 32);
    }
    const float i21 = 1.0f / 21.0f;
    float cw = wing_abs(sx * i21) + wing_abs(sy * i21) + wing_abs(sz * i21);

    float contrib = (0.4f / 441.0f) * covsum + (0.6f / 66.0f) * psum;
    if (lane == 0) contrib += (0.6f / 66.0f) * cw;      // add centroid row once
    acc += contrib;
  }

  // Wave reduce, then block reduce -> one partial per block (deterministic).
#pragma unroll
  for (int m = 16; m > 0; m >>= 1) acc += __shfl_xor(acc, m, 32);

  __shared__ float wsum[8];
  if (lane == 0) wsum[widx] = acc;
  __syncthreads();
  if (threadIdx.x == 0) {
    float t = 0.0f;
    for (int i = 0; i < nwaveblk; ++i) t += wsum[i];
    partial[blockIdx.x] = t;
  }
}

__global__ void reduce_partials(const float* __restrict__ p, int n,
                                float* __restrict__ out, float scale) {
  __shared__ float sm[256];
  float t = 0.0f;
  for (int i = threadIdx.x; i < n; i += blockDim.x) t += p[i];
  sm[threadIdx.x] = t;
  __syncthreads();
  for (int s = 128; s > 0; s >>= 1) {
    if ((int)threadIdx.x < s) sm[threadIdx.x] += sm[threadIdx.x + s];
    __syncthreads();
  }
  if (threadIdx.x == 0) out[0] = sm[0] * scale;
}

extern "C" void kernel_launch(void* const* d_in, const int* in_sizes, int n_in,
                              void* d_out, int out_size, void* d_ws, size_t ws_size,
                              hipStream_t stream) {
  const float* jo = (const float*)d_in[0];
  const float* jg = (const float*)d_in[1];
  float* out      = (float*)d_out;
  float* partial  = (float*)d_ws;

  const int nbatch = in_sizes[0] / 63;

  int blocks = 2048;
  if ((size_t)blocks * sizeof(float) > ws_size) {
    blocks = (int)(ws_size / sizeof(float));
    if (blocks < 1) blocks = 1;
  }

  wing_total_kernel<<<blocks, 256, 0, stream>>>(jo, jg, partial, nbatch);
  reduce_partials<<<1, 256, 0, stream>>>(partial, blocks, out,
                                         1.0f / (float)nbatch);
}